// VectorQuantizerEMA_27298812133947
// MI455X (gfx1250) — compile-verified
//
#include <hip/hip_runtime.h>
#include <math.h>

typedef __attribute__((ext_vector_type(2))) float v2f;
typedef __attribute__((ext_vector_type(4))) float v4f;
typedef __attribute__((ext_vector_type(8))) float v8f;

// Problem constants
#define BDIM 16
#define DDIM 64
#define TDIM 4096
#define KDIM 512
#define NROWS (BDIM * TDIM)            // 65536
#define ROWS_PER_BLOCK 128
#define NBLOCKS (NROWS / ROWS_PER_BLOCK) // 512

// d_out layout (floats): [0]=loss, [1..4194304]=quantized (B,D,T),
// [4194305]=perplexity, [4194306..]=encodings (N,K)
#define QUANT_OFF 1
#define PPL_OFF   (1 + BDIM * DDIM * TDIM)          // 4194305
#define ENC_OFF   (2 + BDIM * DDIM * TDIM)          // 4194306

// d_ws layout: f32[0..511]=enorm, u32[512..1023]=hist, f32[1024..1535]=loss partials
#define WS_HIST 512
#define WS_LOSS 1024

#define LDSX_STRIDE 136   // floats; 2*136 mod 64 == 16 -> conflict-free A-frag reads

// ---------------- Kernel 1: enorm + zero accumulators -----------------------
__global__ __launch_bounds__(512) void vq_prep(const float* __restrict__ E,
                                               float* __restrict__ ws) {
    const int k = threadIdx.x;          // 512 threads, one per codeword
    const float* row = E + k * DDIM;
    float s = 0.f;
#pragma unroll
    for (int d = 0; d < DDIM; ++d) { float v = row[d]; s += v * v; }
    ws[k] = s;
    ((unsigned int*)ws)[WS_HIST + k] = 0u;
    ws[WS_LOSS + k] = 0.f;
}

// ---------------- Kernel 2: distances via WMMA, argmin, outputs -------------
__global__ __launch_bounds__(256) void vq_main(const float* __restrict__ x,
                                               const float* __restrict__ E,
                                               const float* __restrict__ ws,
                                               unsigned int* __restrict__ hist,
                                               float* __restrict__ lossPartial,
                                               float* __restrict__ out) {
    __shared__ float ldsX[DDIM * LDSX_STRIDE];  // [d][tt], tt in 0..127
    __shared__ float ldsEn[KDIM];
    __shared__ int   ldsIdx[ROWS_PER_BLOCK];
    __shared__ float ldsMin[ROWS_PER_BLOCK];
    __shared__ float ldsXn[ROWS_PER_BLOCK];
    __shared__ float ldsLoss[ROWS_PER_BLOCK];

    const int tid = threadIdx.x;
    const int blk = blockIdx.x;
    const int b   = blk >> 5;            // 32 blocks per batch image
    const int t0  = (blk & 31) * ROWS_PER_BLOCK;
    const int n0  = blk * ROWS_PER_BLOCK;

    // stage ||e_k||^2
    ldsEn[tid]       = ws[tid];
    ldsEn[tid + 256] = ws[tid + 256];

    // Load X tile coalesced along t (float4), store [d][tt]
    const float* xbase = x + (size_t)b * (DDIM * TDIM) + t0;
#pragma unroll
    for (int it = 0; it < 8; ++it) {
        int i4  = tid + 256 * it;        // 0..2047 float4s
        int d   = i4 >> 5;               // 32 float4 per d-row
        int tt4 = i4 & 31;
        v4f v = *(const v4f*)(xbase + (size_t)d * TDIM + tt4 * 4);
        *(v4f*)(&ldsX[d * LDSX_STRIDE + tt4 * 4]) = v;
    }
    __syncthreads();

    const int wave = tid >> 5;
    const int lane = tid & 31;
    const int hh   = lane >> 4;          // half-wave id
    const int lr   = lane & 15;
    const int rowt = wave * 16 + lr;     // A-matrix row (tt) for this lane

    // A fragments: a[i] holds X[rowt][4i + 2*hh], X[rowt][4i + 2*hh + 1]
    v2f a[16];
#pragma unroll
    for (int i = 0; i < 16; ++i) {
        int k = 4 * i + 2 * hh;
        a[i].x = ldsX[k * LDSX_STRIDE + rowt];
        a[i].y = ldsX[(k + 1) * LDSX_STRIDE + rowt];
    }

    // ||x_row||^2 (this lane holds 32 of 64 elements; partner half has the rest)
    float xnp = 0.f;
#pragma unroll
    for (int i = 0; i < 16; ++i) xnp += a[i].x * a[i].x + a[i].y * a[i].y;
    float xn = xnp + __shfl_xor(xnp, 16, 32);

    float best[8];
    int   bidx[8];
#pragma unroll
    for (int r = 0; r < 8; ++r) { best[r] = 3.4e38f; bidx[r] = 0x7fffffff; }

    for (int ct = 0; ct < KDIM / 16; ++ct) {
        const int colBase = ct * 16;
        const float* erow = E + (size_t)(colBase + lr) * DDIM + 2 * hh;
        // Two independent accumulator chains to break the WMMA RAW chain
        v8f c0 = {};
        v8f c1 = {};
#pragma unroll
        for (int i = 0; i < 8; ++i) {
            v2f bf0 = *(const v2f*)(erow + 4 * i);            // K = 4i..
            v2f bf1 = *(const v2f*)(erow + 4 * (i + 8));      // K = 4i+32..
            c0 = __builtin_amdgcn_wmma_f32_16x16x4_f32(
                     false, a[i], false, bf0, (short)0, c0, false, false);
            c1 = __builtin_amdgcn_wmma_f32_16x16x4_f32(
                     false, a[i + 8], false, bf1, (short)0, c1, false, false);
        }
        const float en = ldsEn[colBase + lr];
        const int   col = colBase + lr;
#pragma unroll
        for (int r = 0; r < 8; ++r) {
            float dot = c0[r] + c1[r];
            float dist = fmaf(-2.f, dot, en);    // ||e||^2 - 2 x.e  (row-const dropped)
            if (dist < best[r]) { best[r] = dist; bidx[r] = col; } // cols ascend -> first-min
        }
    }

    // min-reduce (with first-index tie break) across the 16 columns in each half
#pragma unroll
    for (int m = 1; m < 16; m <<= 1) {
#pragma unroll
        for (int r = 0; r < 8; ++r) {
            float ov = __shfl_xor(best[r], m, 32);
            int   oi = __shfl_xor(bidx[r], m, 32);
            if (ov < best[r] || (ov == best[r] && oi < bidx[r])) {
                best[r] = ov; bidx[r] = oi;
            }
        }
    }
    if (lr == 0) {
#pragma unroll
        for (int r = 0; r < 8; ++r) {
            int row = wave * 16 + hh * 8 + r;     // C layout: M = r + 8*half
            ldsMin[row] = best[r];
            ldsIdx[row] = bidx[r];
        }
    }
    if (hh == 0) ldsXn[wave * 16 + lr] = xn;
    __syncthreads();

    // loss contributions + histogram
    if (tid < ROWS_PER_BLOCK) {
        ldsLoss[tid] = ldsXn[tid] + ldsMin[tid];  // == ||x - e_idx||^2
        atomicAdd(&hist[ldsIdx[tid]], 1u);
    }
    __syncthreads();
    if (tid == 0) {
        float s = 0.f;
        for (int i = 0; i < ROWS_PER_BLOCK; ++i) s += ldsLoss[i];
        lossPartial[blk] = s;                     // overwrite: replay-safe
    }

    // quantized output, written coalesced along t: out[b,d,t] = E[idx[t]][d]
    float* q = out + QUANT_OFF + (size_t)b * (DDIM * TDIM) + t0;
#pragma unroll 4
    for (int it = 0; it < 32; ++it) {
        int i  = tid + 256 * it;                  // 8192 elements
        int d  = i >> 7;
        int tt = i & 127;
        float v = E[(size_t)ldsIdx[tt] * DDIM + d];
        __builtin_nontemporal_store(v, q + (size_t)d * TDIM + tt);
    }

    // one-hot encodings: 128 rows x 512, streamed with NT v2f stores (base is 8B aligned)
    float* enc = out + ENC_OFF + (size_t)n0 * KDIM;
#pragma unroll 4
    for (int it = 0; it < 128; ++it) {
        int i2  = tid + 256 * it;                 // 0..32767 float-pairs
        int c2  = i2 & 255;                       // 256 pairs per row
        int row = i2 >> 8;
        int k   = ldsIdx[row];
        v2f v = {0.f, 0.f};
        if ((k >> 1) == c2) { if (k & 1) v.y = 1.f; else v.x = 1.f; }
        __builtin_nontemporal_store(v, (v2f*)(enc + (size_t)row * KDIM + c2 * 2));
    }
}

// ---------------- Kernel 3: perplexity + loss scalars -----------------------
__global__ __launch_bounds__(512) void vq_final(const float* __restrict__ ws,
                                                float* __restrict__ out) {
    __shared__ float red[KDIM];
    const int k = threadIdx.x;
    const unsigned int* hist = (const unsigned int*)ws + WS_HIST;
    float p = (float)hist[k] * (1.0f / (float)NROWS);
    red[k] = p * logf(p + 1e-10f);
    __syncthreads();
    for (int s = 256; s > 0; s >>= 1) {
        if (k < s) red[k] += red[k + s];
        __syncthreads();
    }
    if (k == 0) {
        out[PPL_OFF] = expf(-red[0]);
        float ls = 0.f;
        const float* lp = ws + WS_LOSS;
        for (int i = 0; i < NBLOCKS; ++i) ls += lp[i];
        out[0] = 0.25f * ls / (float)((size_t)NROWS * DDIM);
    }
}

// ---------------------------------------------------------------------------
extern "C" void kernel_launch(void* const* d_in, const int* in_sizes, int n_in,
                              void* d_out, int out_size, void* d_ws, size_t ws_size,
                              hipStream_t stream) {
    const float* x = (const float*)d_in[0];     // (16, 64, 4096) fp32
    const float* E = (const float*)d_in[1];     // (512, 64) fp32
    float* out = (float*)d_out;
    float* ws  = (float*)d_ws;

    vq_prep<<<1, 512, 0, stream>>>(E, ws);
    vq_main<<<NBLOCKS, 256, 0, stream>>>(x, E, ws,
                                         (unsigned int*)d_ws + WS_HIST,
                                         ws + WS_LOSS, out);
    vq_final<<<1, 512, 0, stream>>>(ws, out);
}